// GNNEncoder_26474178412606
// MI455X (gfx1250) — compile-verified
//
#include <hip/hip_runtime.h>

// GraphSAGE 2-layer forward for MI455X (gfx1250).
// N=100000 nodes, E=1.6M edges, all feature dims = 128 (fp32).
//
// Perf model: feature matrices (51.2 MB) are L2-resident (192 MB L2), so the
// edge gather/scatter-add (819 MB of loads + 204.8M f32 atomics per layer)
// runs at L2 rates and dominates. GEMMs (~13 GFLOP total) are negligible, so
// we use the exact-precision V_WMMA_F32_16X16X4_F32 path: full fp32 numerics
// at effectively zero cost, while exercising the CDNA5 matrix pipe.

#define NF 128  // D_IN == HID == OUT

typedef __attribute__((ext_vector_type(2))) float v2f;
typedef __attribute__((ext_vector_type(8))) float v8f;

__global__ void zero_f_k(float* __restrict__ p, int n) {
    int i = blockIdx.x * blockDim.x + threadIdx.x;
    if (i < n) p[i] = 0.0f;
}

__global__ void zero_f4_k(float4* __restrict__ p, long long n4) {
    long long i = (long long)blockIdx.x * blockDim.x + threadIdx.x;
    if (i < n4) p[i] = make_float4(0.f, 0.f, 0.f, 0.f);
}

// deg[dst] += 1 for every edge (f32 atomics; deg values are small exact ints)
__global__ void degree_k(const long long* __restrict__ dst,
                         float* __restrict__ deg, long long nE) {
    long long e = (long long)blockIdx.x * blockDim.x + threadIdx.x;
    if (e < nE) atomicAdd(&deg[(int)dst[e]], 1.0f);
}

__global__ void invdeg_k(float* __restrict__ deg, int n) {
    int i = blockIdx.x * blockDim.x + threadIdx.x;
    if (i < n) {
        float d = deg[i];
        deg[i] = (d > 0.f) ? (1.0f / fmaxf(d, 1.0f)) : 0.f;
    }
}

// One wave per edge: contiguous 512B float4 gather of feat[src], 128 f32
// atomic adds into sum[dst]. Both sides are 51.2 MB buffers -> L2 resident.
__global__ void scatter_k(const float* __restrict__ feat,
                          const long long* __restrict__ src,
                          const long long* __restrict__ dst,
                          float* __restrict__ sum, long long nE) {
    long long tid = (long long)blockIdx.x * blockDim.x + threadIdx.x;
    long long e = tid >> 5;          // 32 lanes cover one edge's 128 features
    int c = (int)(tid & 31);         // float4 chunk 0..31
    if (e >= nE) return;
    int s = (int)src[e];
    int d = (int)dst[e];
    float4 v = ((const float4*)(feat + (long long)s * NF))[c];
    float* p = sum + (long long)d * NF + (c << 2);
    atomicAdd(p + 0, v.x);
    atomicAdd(p + 1, v.y);
    atomicAdd(p + 2, v.z);
    atomicAdd(p + 3, v.w);
}

// Fused SAGEConv linear stage:
//   out = act( (sum * inv_deg) @ Wl^T + bl + xroot @ Wr^T )
// Block = 16-node tile; 8 waves, wave w owns output columns [16w, 16w+16).
// A tiles staged through LDS (coalesced float4), agg pre-scaled by inv_deg.
// Both products accumulated with V_WMMA_F32_16X16X4_F32 (exact fp32).
__global__ __launch_bounds__(256)
void sage_gemm_k(const float* __restrict__ aggsum,
                 const float* __restrict__ xroot,
                 const float* __restrict__ invdeg,
                 const float* __restrict__ Wl,   // [128,128] row-major
                 const float* __restrict__ bl,   // [128]
                 const float* __restrict__ Wr,   // [128,128] row-major
                 float* __restrict__ out,
                 int n, int do_relu) {
    __shared__ float sA[16][NF + 4];  // inv_deg-scaled aggregate tile
    __shared__ float sX[16][NF + 4];  // root-feature tile

    const int row0 = blockIdx.x * 16;

    // Cooperative coalesced load of both 16x128 A-tiles (512 float4s each).
    for (int i = threadIdx.x; i < 16 * (NF / 4); i += blockDim.x) {
        int rr = i >> 5;                      // tile row 0..15
        int cc = i & 31;                      // float4 column 0..31
        int row = min(row0 + rr, n - 1);      // clamp tail (duplicates, unused)
        float sc = invdeg[row];
        float4 va = ((const float4*)(aggsum + (long long)row * NF))[cc];
        float4 vx = ((const float4*)(xroot + (long long)row * NF))[cc];
        sA[rr][cc * 4 + 0] = va.x * sc;
        sA[rr][cc * 4 + 1] = va.y * sc;
        sA[rr][cc * 4 + 2] = va.z * sc;
        sA[rr][cc * 4 + 3] = va.w * sc;
        *(float4*)&sX[rr][cc * 4] = vx;
    }
    __syncthreads();

    const int wave = threadIdx.x >> 5;   // 0..7 -> output column block
    const int lane = threadIdx.x & 31;
    const int col0 = wave * 16;
    const int r  = lane & 15;            // A row / B column within tile
    const int kq = (lane >> 4) << 1;     // K sub-offset: 0 (lo half) / 2 (hi)

    // B fragment: B[k][c] = W[c][k] (x @ W^T), same 2-float stride pattern.
    const float* bL = Wl + (long long)(col0 + r) * NF + kq;
    const float* bR = Wr + (long long)(col0 + r) * NF + kq;

    v8f accL = {};  // (agg*invdeg) @ Wl^T
    v8f accR = {};  // xroot @ Wr^T
#pragma unroll
    for (int k = 0; k < NF; k += 4) {
        v2f aL = *(const v2f*)&sA[r][k + kq];
        v2f aX = *(const v2f*)&sX[r][k + kq];
        v2f wl = *(const v2f*)(bL + k);
        v2f wr = *(const v2f*)(bR + k);
        // (neg_a, A, neg_b, B, c_mod, C, reuse_a, reuse_b)
        accL = __builtin_amdgcn_wmma_f32_16x16x4_f32(
            false, aL, false, wl, (short)0, accL, false, false);
        accR = __builtin_amdgcn_wmma_f32_16x16x4_f32(
            false, aX, false, wr, (short)0, accR, false, false);
    }

    // C/D layout: VGPR i -> row i (lanes 0-15) or row 8+i (lanes 16-31),
    // column = lane & 15.
    const float bias = bl[col0 + r];
    const int half = (lane >> 4) * 8;
#pragma unroll
    for (int i = 0; i < 8; ++i) {
        int row = row0 + half + i;
        if (row < n) {
            float v = accL[i] + accR[i] + bias;
            if (do_relu) v = fmaxf(v, 0.f);
            out[(long long)row * NF + col0 + r] = v;
        }
    }
}

extern "C" void kernel_launch(void* const* d_in, const int* in_sizes, int n_in,
                              void* d_out, int out_size, void* d_ws, size_t ws_size,
                              hipStream_t stream) {
    const float*     x   = (const float*)d_in[0];
    const long long* ei  = (const long long*)d_in[1];  // int64 (2,E)
    const float*     W1l = (const float*)d_in[2];
    const float*     b1l = (const float*)d_in[3];
    const float*     W1r = (const float*)d_in[4];
    const float*     W2l = (const float*)d_in[5];
    const float*     b2l = (const float*)d_in[6];
    const float*     W2r = (const float*)d_in[7];
    float* out = (float*)d_out;

    const int       n  = in_sizes[0] / NF;       // 100000
    const long long nE = in_sizes[1] / 2;        // 1.6M
    const long long* src = ei;
    const long long* dst = ei + nE;

    // Workspace: inv_deg [n], sum [n*128], h [n*128]  (~103 MB)
    float* invdeg = (float*)d_ws;
    size_t npad = ((size_t)n + 1023) & ~(size_t)1023;
    float* sum = invdeg + npad;
    float* h   = sum + (size_t)n * NF;
    if (ws_size < (npad + 2ull * (size_t)n * NF) * sizeof(float)) return;

    const int BT = 256;
    const long long nSum4  = (long long)n * NF / 4;
    const long long nScat  = nE * 32;
    const unsigned gZeroS  = (unsigned)((nSum4 + BT - 1) / BT);
    const unsigned gDeg    = (unsigned)((nE + BT - 1) / BT);
    const unsigned gNode   = (unsigned)((n + BT - 1) / BT);
    const unsigned gScat   = (unsigned)((nScat + BT - 1) / BT);
    const unsigned gGemm   = (unsigned)((n + 15) / 16);

    // degrees -> inv_deg
    zero_f_k<<<gNode, BT, 0, stream>>>(invdeg, n);
    degree_k<<<gDeg, BT, 0, stream>>>(dst, invdeg, nE);
    invdeg_k<<<gNode, BT, 0, stream>>>(invdeg, n);

    // layer 1: aggregate x, then h = relu(agg@W1l^T + b1l + x@W1r^T)
    zero_f4_k<<<gZeroS, BT, 0, stream>>>((float4*)sum, nSum4);
    scatter_k<<<gScat, BT, 0, stream>>>(x, src, dst, sum, nE);
    sage_gemm_k<<<gGemm, BT, 0, stream>>>(sum, x, invdeg, W1l, b1l, W1r, h, n, 1);

    // layer 2: aggregate h, then out = agg@W2l^T + b2l + h@W2r^T
    zero_f4_k<<<gZeroS, BT, 0, stream>>>((float4*)sum, nSum4);
    scatter_k<<<gScat, BT, 0, stream>>>(h, src, dst, sum, nE);
    sage_gemm_k<<<gGemm, BT, 0, stream>>>(sum, h, invdeg, W2l, b2l, W2r, out, n, 0);
}